// Lenet5_54915451847260
// MI455X (gfx1250) — compile-verified
//
#include <hip/hip_runtime.h>
#include <cstdint>

typedef float v4f __attribute__((ext_vector_type(4)));

static constexpr int kH = 256, kW = 256;   // input spatial dims
static constexpr int kOH = 128, kOW = 128; // output spatial dims
static constexpr int kWavesPerBlock = 8;   // 256 threads, wave32

// One wave32 per output row:
//  - stage the two source input rows (2048 contiguous bytes) into LDS with
//    four global_load_async_to_lds_b128 (ASYNCcnt-tracked, NT hint: data is
//    read exactly once, don't pollute L2)
//  - s_wait_asynccnt 0 (wave-local; no barrier needed), ds_load_b128 reads
//  - 2x2 sum, sigmoid via v_exp_f32 + v_rcp_f32 (1-ulp rcp is plenty for a
//    sigmoid output; avoids the ~7-op correctly-rounded divide expansion)
//  - one NT b128 store per lane (512 B contiguous per wave)
__global__ __launch_bounds__(256) void pool2x2_sigmoid_async(
    const float* __restrict__ x,
    const float* __restrict__ coef,
    const float* __restrict__ bias,
    float* __restrict__ out) {
  __shared__ float smem[kWavesPerBlock * 2 * kW];  // 8 waves * 2 rows * 256 f32 = 16 KB

  const int tid  = (int)threadIdx.x;
  const int wave = tid >> 5;
  const int lane = tid & 31;

  const unsigned r  = blockIdx.x * kWavesPerBlock + wave;  // output row id [0, 524288)
  const unsigned bc = r >> 7;                              // fused (b,c) index
  const unsigned oh = r & (kOH - 1);

  const float cf = coef[0];
  const float bs = bias[0];

  // Two consecutive input rows = 2048 contiguous bytes in HBM and in LDS.
  const float* gbase = x + (size_t)bc * (size_t)(kH * kW) + (size_t)(2u * oh) * (size_t)kW;
  float* sbase = &smem[wave * 2 * kW];

  // Per-lane addresses: 16 bytes per lane per instruction.
  uint64_t gaddr = (uint64_t)(uintptr_t)(gbase + 4 * lane);
  uint32_t laddr = (uint32_t)(uintptr_t)(sbase + 4 * lane);  // low 32 bits = LDS byte offset

  // INST_OFFSET is added to BOTH the global and LDS addresses (ISA 08 section 4.4),
  // so one base pair + 4 offsets streams the whole 2 KB row-pair.
  asm volatile("global_load_async_to_lds_b128 %0, %1, off th:TH_LOAD_NT"             :: "v"(laddr), "v"(gaddr) : "memory");
  asm volatile("global_load_async_to_lds_b128 %0, %1, off offset:512 th:TH_LOAD_NT"  :: "v"(laddr), "v"(gaddr) : "memory");
  asm volatile("global_load_async_to_lds_b128 %0, %1, off offset:1024 th:TH_LOAD_NT" :: "v"(laddr), "v"(gaddr) : "memory");
  asm volatile("global_load_async_to_lds_b128 %0, %1, off offset:1536 th:TH_LOAD_NT" :: "v"(laddr), "v"(gaddr) : "memory");
  asm volatile("s_wait_asynccnt 0x0" ::: "memory");

  // Each lane consumes input columns [8*lane, 8*lane+7] of both rows ->
  // output columns [4*lane, 4*lane+3].
  const v4f a = *(const v4f*)(sbase + 8 * lane);            // row0 cols 8l..8l+3
  const v4f b = *(const v4f*)(sbase + 8 * lane + 4);        // row0 cols 8l+4..8l+7
  const v4f c = *(const v4f*)(sbase + kW + 8 * lane);       // row1 cols 8l..8l+3
  const v4f d = *(const v4f*)(sbase + kW + 8 * lane + 4);   // row1 cols 8l+4..8l+7

  const float p0 = (a.x + a.y) + (c.x + c.y);
  const float p1 = (a.z + a.w) + (c.z + c.w);
  const float p2 = (b.x + b.y) + (d.x + d.y);
  const float p3 = (b.z + b.w) + (d.z + d.w);

  v4f o;
  o.x = __builtin_amdgcn_rcpf(1.0f + __expf(-__builtin_fmaf(cf, p0, bs)));
  o.y = __builtin_amdgcn_rcpf(1.0f + __expf(-__builtin_fmaf(cf, p1, bs)));
  o.z = __builtin_amdgcn_rcpf(1.0f + __expf(-__builtin_fmaf(cf, p2, bs)));
  o.w = __builtin_amdgcn_rcpf(1.0f + __expf(-__builtin_fmaf(cf, p3, bs)));

  // Output is write-once: non-temporal 128-bit store, 512 B contiguous per wave.
  __builtin_nontemporal_store(o, (v4f*)(out + (size_t)r * kOW + 4 * lane));
}

extern "C" void kernel_launch(void* const* d_in, const int* in_sizes, int n_in,
                              void* d_out, int out_size, void* d_ws, size_t ws_size,
                              hipStream_t stream) {
  (void)in_sizes; (void)n_in; (void)out_size; (void)d_ws; (void)ws_size;
  const float* x    = (const float*)d_in[0];
  const float* coef = (const float*)d_in[1];
  const float* bias = (const float*)d_in[2];
  float* out = (float*)d_out;

  // Output rows: B*C*OH = 64*64*128 = 524288; 8 rows per block -> 65536 blocks.
  const int total_rows = 64 * 64 * kOH;
  const int blocks = total_rows / kWavesPerBlock;
  pool2x2_sigmoid_async<<<blocks, 256, 0, stream>>>(x, coef, bias, out);
}